// Conv3x3_72516227826027
// MI455X (gfx1250) — compile-verified
//
#include <hip/hip_runtime.h>

// out[pw*2048+ph] = -(1/0.924458) * sum_{i,j} w[3i+j] * x[3pw+i][3ph+j]
// x: 6144x6144 f32, w: 9 f32, out: 2048*2048 f32.
// HBM-bound kernel; dot products done on the f32 WMMA pipe (16x16x4_f32, K=9 padded to 12).

typedef float v2f __attribute__((ext_vector_type(2)));
typedef float v8f __attribute__((ext_vector_type(8)));

#define XDIM   6144
#define NPH    2048
#define NPAIRS (2048 * 64)          // each "pair" = 32 patches (2 WMMA tiles of 16)
#define SCALE  (-1.0f / 0.924458f)  // folded readout gain: -q2/c1 per unit C_BASE

__global__ __launch_bounds__(256) void conv3x3_wmma(const float* __restrict__ x,
                                                    const float* __restrict__ w,
                                                    float* __restrict__ out)
{
    __shared__ float xch[8 * 32];               // 32-float exchange slot per wave
    const int  lane = threadIdx.x & 31;
    const int  wave = threadIdx.x >> 5;
    const int  gw   = blockIdx.x * 8 + wave;
    const int  nwav = gridDim.x * 8;
    const bool lo   = lane < 16;
    const int  m    = lane & 15;                // patch row (M) handled by this lane

    // Pre-scaled taps.
    const float w0 = w[0] * SCALE, w1 = w[1] * SCALE, w2 = w[2] * SCALE,
                w3 = w[3] * SCALE, w4 = w[4] * SCALE, w5 = w[5] * SCALE,
                w6 = w[6] * SCALE, w7 = w[7] * SCALE, w8 = w[8] * SCALE;

    // B (4x16 f32) fragments, filter broadcast to all 16 columns.
    // Layout: VGPR0 = {K0 | K2}, VGPR1 = {K1 | K3} across lane halves.
    const v2f B0 = { lo ? w0 : w2, lo ? w1 : w3 };   // k = 0..3
    const v2f B1 = { lo ? w4 : w6, lo ? w5 : w7 };   // k = 4..7
    const v2f B2 = { lo ? w8 : 0.0f, 0.0f };         // k = 8 (9..11 zero pad)

    // One 16-patch tile: gather A fragments straight from global, 3 chained WMMAs.
    auto tileC = [&](const float* g) -> v8f {
        // lo lanes need k0,k1,k4,k5,k8 ; hi lanes need k2,k3,k6,k7 (k = 3*i + j).
        float f0 = g[lo ? 0        : 2];            // k0 | k2
        float f1 = g[lo ? 1        : XDIM];         // k1 | k3
        float f2 = g[lo ? XDIM + 1 : 2 * XDIM];     // k4 | k6
        float f3 = g[lo ? XDIM + 2 : 2 * XDIM + 1]; // k5 | k7
        float f4 = g[2 * XDIM + 2];                 // k8 (hi half discards)
        v2f A0 = { f0, f1 };
        v2f A1 = { f2, f3 };
        v2f A2 = { lo ? f4 : 0.0f, 0.0f };
        v8f c = {};
        c = __builtin_amdgcn_wmma_f32_16x16x4_f32(false, A0, false, B0, (short)0, c, false, false);
        c = __builtin_amdgcn_wmma_f32_16x16x4_f32(false, A1, false, B1, (short)0, c, false, false);
        c = __builtin_amdgcn_wmma_f32_16x16x4_f32(false, A2, false, B2, (short)0, c, false, false);
        return c;
    };

    // C layout: VGPR r holds M=r (lanes 0-15) / M=r+8 (lanes 16-31), all N equal.
    auto pick = [](v8f c, int idx) -> float {
        float v = c[0];
        v = (idx == 1) ? c[1] : v;
        v = (idx == 2) ? c[2] : v;
        v = (idx == 3) ? c[3] : v;
        v = (idx == 4) ? c[4] : v;
        v = (idx == 5) ? c[5] : v;
        v = (idx == 6) ? c[6] : v;
        v = (idx == 7) ? c[7] : v;
        return v;
    };

    for (int pr = gw; pr < NPAIRS; pr += nwav) {
        const int pw     = pr >> 6;
        const int phBase = (pr & 63) << 5;
        const float* g0  = x + (size_t)(3 * pw) * XDIM + 3 * phBase + 3 * m;

        v8f cA = tileC(g0);        // patches phBase .. phBase+15
        v8f cB = tileC(g0 + 48);   // patches phBase+16 .. phBase+31

        // Collapse replicated columns -> 32 contiguous outputs via per-wave LDS slot.
        const int idx = lane & 7;
        float vA = pick(cA, idx);
        float vB = pick(cB, idx);
        if ((lane & 15) < 8) {     // one writer per output: lanes 0-7 (M 0-7), 16-23 (M 8-15)
            int wa = wave * 32 + ((lane & 7) | ((lane >> 1) & 8));
            xch[wa]      = vA;
            xch[wa + 16] = vB;
        }
        __builtin_amdgcn_wave_barrier();   // DS ops are in-order within a wave
        float r = xch[wave * 32 + lane];
        out[(size_t)pw * NPH + phBase + lane] = r;   // 128B coalesced store per wave
    }
}

extern "C" void kernel_launch(void* const* d_in, const int* in_sizes, int n_in,
                              void* d_out, int out_size, void* d_ws, size_t ws_size,
                              hipStream_t stream)
{
    const float* x = (const float*)d_in[0];   // 6144*6144
    const float* w = (const float*)d_in[1];   // 9
    float* out = (float*)d_out;               // 2048*2048
    (void)in_sizes; (void)n_in; (void)out_size; (void)d_ws; (void)ws_size;
    // 4096 blocks * 8 waves = 32768 waves; 131072 pairs -> exactly 4 per wave.
    hipLaunchKernelGGL(conv3x3_wmma, dim3(4096), dim3(256), 0, stream, x, w, out);
}